// FeaturePropagation_81913616270005
// MI455X (gfx1250) — compile-verified
//
#include <hip/hip_runtime.h>

#define K_NN   3
#define EPS_W  1e-8f
#define BN_EPS 1e-5f
#define BATCH  4
#define NYQ    16384
#define NXP    4096
#define CYF    128
#define CXF    256
#define DIMF   384            // CYF + CXF
#define MTOT   (BATCH * NYQ)  // 65536 rows total

#define KT     32             // K tile per LDS stage
#define AST    36             // padded LDS row stride (floats): 144B, 16B-aligned,
                              // bank-conflict-free for the 16-row fragment reads

typedef float v2f __attribute__((ext_vector_type(2)));
typedef float v8f __attribute__((ext_vector_type(8)));

// ---------------------------------------------------------------------------
// Kernel 1: 3-NN search + inverse-distance weights.
// One batch's x_points (4096 x 3 = 48KB) staged in LDS; each thread scans all
// candidates for one query point, keeping the 3 smallest squared distances.
// ---------------------------------------------------------------------------
__global__ void knn_kernel(const float* __restrict__ ypts,
                           const float* __restrict__ xpts,
                           int* __restrict__ idx_out,
                           float* __restrict__ w_out) {
    __shared__ float sx[NXP * 3];
    const int b = blockIdx.y;
    const float* xb = xpts + (size_t)b * NXP * 3;
    for (int t = threadIdx.x; t < NXP * 3; t += blockDim.x) sx[t] = xb[t];
    __syncthreads();

    const int n = blockIdx.x * blockDim.x + threadIdx.x;
    const float* yp = ypts + ((size_t)b * NYQ + n) * 3;
    const float y0 = yp[0], y1 = yp[1], y2 = yp[2];
    const float ysq = y0 * y0 + y1 * y1 + y2 * y2;

    float d0 = INFINITY, d1 = INFINITY, d2 = INFINITY;
    int i0 = 0, i1 = 0, i2 = 0;
    for (int j = 0; j < NXP; ++j) {
        const float x0 = sx[3 * j + 0];
        const float x1 = sx[3 * j + 1];
        const float x2 = sx[3 * j + 2];
        const float xsq = x0 * x0 + x1 * x1 + x2 * x2;
        const float dot = y0 * x0 + y1 * x1 + y2 * x2;
        const float d = ysq + xsq - 2.0f * dot;
        if (d < d0)      { d2 = d1; i2 = i1; d1 = d0; i1 = i0; d0 = d; i0 = j; }
        else if (d < d1) { d2 = d1; i2 = i1; d1 = d;  i1 = j; }
        else if (d < d2) { d2 = d;  i2 = j; }
    }
    float w0 = 1.0f / (d0 + EPS_W);
    float w1 = 1.0f / (d1 + EPS_W);
    float w2 = 1.0f / (d2 + EPS_W);
    const float inv = 1.0f / (w0 + w1 + w2);
    const size_t o = ((size_t)b * NYQ + n) * 3;
    idx_out[o + 0] = i0; idx_out[o + 1] = i1; idx_out[o + 2] = i2;
    w_out[o + 0] = w0 * inv; w_out[o + 1] = w1 * inv; w_out[o + 2] = w2 * inv;
}

// ---------------------------------------------------------------------------
// Kernel 2: build F[m, 0:128] = y_feats, F[m, 128:384] = weighted neighbor sum
// ---------------------------------------------------------------------------
__global__ void build_f_kernel(const float* __restrict__ yf,
                               const float* __restrict__ xf,
                               const int* __restrict__ idx,
                               const float* __restrict__ w,
                               float* __restrict__ F) {
    const int m = blockIdx.x;
    const int t = threadIdx.x;
    const int b = m / NYQ, n = m % NYQ;
    float v;
    if (t < CYF) {
        v = yf[((size_t)b * NYQ + n) * CYF + t];
    } else {
        const int c = t - CYF;
        const size_t o = (size_t)m * 3;
        const float* xb = xf + (size_t)b * NXP * CXF;
        v = w[o + 0] * xb[(size_t)idx[o + 0] * CXF + c]
          + w[o + 1] * xb[(size_t)idx[o + 1] * CXF + c]
          + w[o + 2] * xb[(size_t)idx[o + 2] * CXF + c];
    }
    F[(size_t)m * DIMF + t] = v;
}

// ---------------------------------------------------------------------------
// Kernel 3: f32 GEMM + bias via V_WMMA_F32_16X16X4_F32 (wave32) with
// double-buffered LDS tiles filled by async global->LDS DMA (ASYNCcnt).
// Y[m, o] = sum_k A[m, k] * W[o, k] + bias[o]
// Block = 256 thr = 8 waves; block tile 128(M) x 64(N), K staged in 32-chunks.
// Each wave computes a 16x64 strip (4 v8f accumulators).
// ---------------------------------------------------------------------------
__device__ __forceinline__ void async_b128_to_lds(const float* gsrc, const float* ldst) {
    const unsigned long long ga = (unsigned long long)(uintptr_t)gsrc;
    const unsigned la = (unsigned)(uintptr_t)ldst;  // low 32 bits = LDS byte offset
    asm volatile("global_load_async_to_lds_b128 %0, %1, off"
                 :: "v"(la), "v"(ga) : "memory");
}

__global__ void __launch_bounds__(256)
gemm_wmma_kernel(const float* __restrict__ A,
                 const float* __restrict__ W,
                 const float* __restrict__ bias,
                 float* __restrict__ Y,
                 int Kd, int Cout) {
    __shared__ float As[2][128 * AST];   // 36 KB
    __shared__ float Bs[2][64 * AST];    // 18 KB

    const int tid  = threadIdx.x;
    const int wave = tid >> 5;
    const int lane = tid & 31;
    const int half = lane >> 4;
    const int lr   = lane & 15;
    const int m0 = blockIdx.x * 128;
    const int n0 = blockIdx.y * 64;

    // Cooperative tile fetch: float4 quanta; 8 quanta per 32-float row.
    // A tile: 128 rows -> 1024 quanta -> 4 per thread.
    // B tile:  64 rows ->  512 quanta -> 2 per thread.
    auto issue_tile = [&](int k0, int buf) {
#pragma unroll
        for (int it = 0; it < 4; ++it) {
            const int q = tid + it * 256;
            const int row = q >> 3, c4 = (q & 7) << 2;
            async_b128_to_lds(A + (size_t)(m0 + row) * Kd + k0 + c4,
                              &As[buf][row * AST + c4]);
        }
#pragma unroll
        for (int it = 0; it < 2; ++it) {
            const int q = tid + it * 256;
            const int row = q >> 3, c4 = (q & 7) << 2;
            async_b128_to_lds(W + (size_t)(n0 + row) * Kd + k0 + c4,
                              &Bs[buf][row * AST + c4]);
        }
    };

    v8f acc0 = {}, acc1 = {}, acc2 = {}, acc3 = {};

    const int T = Kd / KT;
    issue_tile(0, 0);

    for (int t = 0; t < T; ++t) {
        const int buf = t & 1;
        if (t + 1 < T) {
            issue_tile((t + 1) * KT, buf ^ 1);
            // 6 newest (next tile) may stay in flight; 6 oldest (this tile)
            // must retire. Async loads complete in order.
            asm volatile("s_wait_asynccnt 0x6" ::: "memory");
        } else {
            asm volatile("s_wait_asynccnt 0x0" ::: "memory");
        }
        __syncthreads();   // tile published to all waves

        // A frag (16x4): lane<16 row=lr K={k,k+1}; lane>=16 row=lr K={k+2,k+3}
        // B frag (4x16): VGPR r holds K=r (lanes 0-15), K=r+2 (lanes 16-31)
        const float* Ab = &As[buf][(wave * 16 + lr) * AST + 2 * half];
        const float* Bb = &Bs[buf][lr * AST + 2 * half];
#pragma unroll
        for (int k = 0; k < KT; k += 4) {
            const v2f a  = *(const v2f*)(Ab + k);
            const v2f b0 = *(const v2f*)(Bb + k);
            const v2f b1 = *(const v2f*)(Bb + 16 * AST + k);
            const v2f b2 = *(const v2f*)(Bb + 32 * AST + k);
            const v2f b3 = *(const v2f*)(Bb + 48 * AST + k);
            acc0 = __builtin_amdgcn_wmma_f32_16x16x4_f32(false, a, false, b0, (short)0, acc0, false, false);
            acc1 = __builtin_amdgcn_wmma_f32_16x16x4_f32(false, a, false, b1, (short)0, acc1, false, false);
            acc2 = __builtin_amdgcn_wmma_f32_16x16x4_f32(false, a, false, b2, (short)0, acc2, false, false);
            acc3 = __builtin_amdgcn_wmma_f32_16x16x4_f32(false, a, false, b3, (short)0, acc3, false, false);
        }
        __syncthreads();   // all waves done with buf before it is refilled
    }

    // C/D layout: VGPR r -> row = m0 + wave*16 + r + 8*half, col = n0 + lr
    const float bb0 = bias[n0 + lr];
    const float bb1 = bias[n0 + 16 + lr];
    const float bb2 = bias[n0 + 32 + lr];
    const float bb3 = bias[n0 + 48 + lr];
    const int row_base = m0 + wave * 16 + 8 * half;
#pragma unroll
    for (int r = 0; r < 8; ++r) {
        float* yrow = Y + (size_t)(row_base + r) * Cout + n0 + lr;
        yrow[0]  = acc0[r] + bb0;
        yrow[16] = acc1[r] + bb1;
        yrow[32] = acc2[r] + bb2;
        yrow[48] = acc3[r] + bb3;
    }
}

// ---------------------------------------------------------------------------
// BatchNorm statistics: stats[0:512] = per-channel sum, stats[512:1024] = sumsq
// ---------------------------------------------------------------------------
__global__ void zero_stats_kernel(float* __restrict__ stats) {
    const int t = blockIdx.x * blockDim.x + threadIdx.x;
    if (t < 1024) stats[t] = 0.0f;
}

__global__ void reduce_stats_kernel(const float* __restrict__ Y,
                                    float* __restrict__ stats, int Cout) {
    const int c = blockIdx.x * blockDim.x + threadIdx.x;   // blockDim = 128
    const int r0 = blockIdx.y * 1024;                       // gridDim.y = 64
    float s = 0.0f, s2 = 0.0f;
    for (int r = r0; r < r0 + 1024; ++r) {
        const float v = Y[(size_t)r * Cout + c];
        s += v; s2 += v * v;
    }
    atomicAdd(&stats[c], s);
    atomicAdd(&stats[512 + c], s2);
}

// In-place BN (biased var) + affine + ReLU.
__global__ void bn_relu_kernel(float* __restrict__ Y,
                               const float* __restrict__ stats,
                               const float* __restrict__ g,
                               const float* __restrict__ be, int Cout) {
    const size_t e = (size_t)blockIdx.x * blockDim.x + threadIdx.x;
    const int c = (int)(e & (size_t)(Cout - 1));
    const float mean = stats[c] * (1.0f / MTOT);
    const float var  = stats[512 + c] * (1.0f / MTOT) - mean * mean;
    float v = (Y[e] - mean) * rsqrtf(var + BN_EPS);
    v = v * g[c] + be[c];
    Y[e] = v > 0.0f ? v : 0.0f;
}

// Final layer: BN + affine + ReLU + transpose to [B, C=128, NY].
__global__ void bn_relu_out_kernel(const float* __restrict__ Y,
                                   const float* __restrict__ stats,
                                   const float* __restrict__ g,
                                   const float* __restrict__ be,
                                   float* __restrict__ out) {
    const int n = blockIdx.x * blockDim.x + threadIdx.x;
    const int c = blockIdx.y;   // 0..127
    const int b = blockIdx.z;   // 0..3
    const float mean = stats[c] * (1.0f / MTOT);
    const float var  = stats[512 + c] * (1.0f / MTOT) - mean * mean;
    const float y = Y[((size_t)b * NYQ + n) * 128 + c];
    float v = (y - mean) * rsqrtf(var + BN_EPS);
    v = v * g[c] + be[c];
    out[((size_t)b * 128 + c) * NYQ + n] = v > 0.0f ? v : 0.0f;
}

// ---------------------------------------------------------------------------
extern "C" void kernel_launch(void* const* d_in, const int* in_sizes, int n_in,
                              void* d_out, int out_size, void* d_ws, size_t ws_size,
                              hipStream_t stream) {
    const float* ypts = (const float*)d_in[0];
    const float* yfeat = (const float*)d_in[1];
    const float* xpts = (const float*)d_in[2];
    const float* xfeat = (const float*)d_in[3];
    const float* W1 = (const float*)d_in[4];
    const float* b1 = (const float*)d_in[5];
    const float* g1 = (const float*)d_in[6];
    const float* be1 = (const float*)d_in[7];
    const float* W2 = (const float*)d_in[8];
    const float* b2 = (const float*)d_in[9];
    const float* g2 = (const float*)d_in[10];
    const float* be2 = (const float*)d_in[11];
    const float* W3 = (const float*)d_in[12];
    const float* b3 = (const float*)d_in[13];
    const float* g3 = (const float*)d_in[14];
    const float* be3 = (const float*)d_in[15];
    float* out = (float*)d_out;

    // Workspace layout (bytes), aliasing later buffers onto dead ones.
    char* ws = (char*)d_ws;
    const size_t idx_off   = 0;
    const size_t idx_sz    = (size_t)MTOT * 3 * sizeof(int);        // 786 KB
    const size_t w_off     = idx_off + idx_sz;
    const size_t w_sz      = (size_t)MTOT * 3 * sizeof(float);      // 786 KB
    const size_t stats_off = w_off + w_sz;
    const size_t stats_sz  = 1024 * sizeof(float);
    const size_t F_off     = stats_off + stats_sz;
    const size_t F_sz      = (size_t)MTOT * DIMF * sizeof(float);   // 100.7 MB
    const size_t Y1_off    = F_off + F_sz;                           // 134.2 MB
    const size_t Y2_off    = F_off;    // reuse F region (F dead after GEMM1)
    const size_t Y3_off    = Y1_off;   // reuse Y1 region (Y1 dead after GEMM2)

    int*   idxb  = (int*)(ws + idx_off);
    float* wb    = (float*)(ws + w_off);
    float* stats = (float*)(ws + stats_off);
    float* F     = (float*)(ws + F_off);
    float* Y1    = (float*)(ws + Y1_off);
    float* Y2    = (float*)(ws + Y2_off);
    float* Y3    = (float*)(ws + Y3_off);

    // 1) KNN + weights
    knn_kernel<<<dim3(NYQ / 256, BATCH), 256, 0, stream>>>(ypts, xpts, idxb, wb);

    // 2) Interpolate + concat -> F [MTOT, 384]
    build_f_kernel<<<dim3(MTOT), 384, 0, stream>>>(yfeat, xfeat, idxb, wb, F);

    // 3) Layer 1: [MTOT,384] x W1^T[384,512] -> Y1, BN+ReLU in place
    zero_stats_kernel<<<4, 256, 0, stream>>>(stats);
    gemm_wmma_kernel<<<dim3(MTOT / 128, 512 / 64), 256, 0, stream>>>(F, W1, b1, Y1, DIMF, 512);
    reduce_stats_kernel<<<dim3(512 / 128, 64), 128, 0, stream>>>(Y1, stats, 512);
    bn_relu_kernel<<<dim3((unsigned)((size_t)MTOT * 512 / 256)), 256, 0, stream>>>(Y1, stats, g1, be1, 512);

    // 4) Layer 2: [MTOT,512] x W2^T[512,256] -> Y2, BN+ReLU in place
    zero_stats_kernel<<<4, 256, 0, stream>>>(stats);
    gemm_wmma_kernel<<<dim3(MTOT / 128, 256 / 64), 256, 0, stream>>>(Y1, W2, b2, Y2, 512, 256);
    reduce_stats_kernel<<<dim3(256 / 128, 64), 128, 0, stream>>>(Y2, stats, 256);
    bn_relu_kernel<<<dim3((unsigned)((size_t)MTOT * 256 / 256)), 256, 0, stream>>>(Y2, stats, g2, be2, 256);

    // 5) Layer 3: [MTOT,256] x W3^T[256,128] -> Y3, then BN+ReLU+transpose out
    zero_stats_kernel<<<4, 256, 0, stream>>>(stats);
    gemm_wmma_kernel<<<dim3(MTOT / 128, 128 / 64), 256, 0, stream>>>(Y2, W3, b3, Y3, 256, 128);
    reduce_stats_kernel<<<dim3(128 / 128, 64), 128, 0, stream>>>(Y3, stats, 128);
    bn_relu_out_kernel<<<dim3(NYQ / 256, 128, BATCH), 256, 0, stream>>>(Y3, stats, g3, be3, out);

    (void)in_sizes; (void)n_in; (void)out_size; (void)ws_size;
}